// MultiLayerGLA_19825569038708
// MI455X (gfx1250) — compile-verified
//
#include <hip/hip_runtime.h>

#define DD     127
#define TD     128
#define TT     4096
#define NTRAIN 4095
#define NLAYER 2
#define CH     16
#define NCHUNK (TT / CH)
#define EPSV   1e-6f
#define LN_EPS 1e-5f

#if defined(__has_builtin)
#  if __has_builtin(__builtin_amdgcn_global_load_async_to_lds_b128) && \
      __has_builtin(__builtin_amdgcn_s_wait_asynccnt)
#    define USE_ASYNC_LDS 1
#  endif
#endif
#ifndef USE_ASYNC_LDS
#  define USE_ASYNC_LDS 0
#endif

typedef float v2f __attribute__((ext_vector_type(2)));
typedef float v8f __attribute__((ext_vector_type(8)));
typedef int   v4i __attribute__((ext_vector_type(4)));
typedef __attribute__((address_space(1))) v4i* gas_v4i_ptr;
typedef __attribute__((address_space(3))) v4i* las_v4i_ptr;

__device__ __forceinline__ v8f wmma_f32(v2f a, v2f b, v8f c) {
  // D = A(16x4,f32) * B(4x16,f32) + C(16x16,f32)  -> v_wmma_f32_16x16x4_f32
  return __builtin_amdgcn_wmma_f32_16x16x4_f32(false, a, false, b, (short)0, c,
                                               false, false);
}

// Stage one 16x128 f32 token chunk (8 KB) into LDS.
// Async path: 512 x b128 DMA transfers, 2 per thread, tracked by ASYNCcnt.
__device__ __forceinline__ void stage_chunk(const float* __restrict__ g,
                                            float* l, int tid) {
#if USE_ASYNC_LDS
  for (int i = tid; i < CH * TD / 4; i += 256) {
    __builtin_amdgcn_global_load_async_to_lds_b128(
        (gas_v4i_ptr)(g + i * 4),
        (las_v4i_ptr)(l + i * 4),
        /*offset=*/0, /*cpol=*/0);
  }
#else
  for (int i = tid; i < CH * TD; i += 256) l[i] = g[i];
#endif
}

__global__ __launch_bounds__(256, 1)
void gla_fused_kernel(const float* __restrict__ xs, const float* __restrict__ ys,
                      const float* __restrict__ qx,
                      const float* __restrict__ Wq, const float* __restrict__ Wk,
                      const float* __restrict__ Wv, const float* __restrict__ Wgw,
                      const float* __restrict__ Wgb, const float* __restrict__ Wo,
                      const float* __restrict__ Ln1w, const float* __restrict__ Ln1b,
                      const float* __restrict__ Ln2w, const float* __restrict__ Ln2b,
                      const float* __restrict__ predw,
                      float* __restrict__ out, float* __restrict__ tok_ws) {
  __shared__ float sS [TD * TD];      // running state  S_kv (f32, 64 KB)
  __shared__ float sSk[TD];           // running state  S_k
  __shared__ float sXc0[CH * TD];     // raw chunk tokens, buffer 0
  __shared__ float sXc1[CH * TD];     // raw chunk tokens, buffer 1
  __shared__ float sXn[CH * TD];      // LN1 output
  __shared__ float sQp[CH * TD];
  __shared__ float sKp[CH * TD];
  __shared__ float sVg[CH * TD];
  __shared__ float sAt[CH * TD];      // attn_out
  __shared__ float sOu[CH * TD];      // pre-LN2 output
  __shared__ float sA [CH * CH];      // masked Qp Kp^T
  __shared__ float sDp[CH * 16];      // denominator partials
  __shared__ float sDen[CH];
  __shared__ float sFin[TD];          // final token row
  __shared__ float sL1w[TD], sL1b[TD], sL2w[TD], sL2b[TD], sGb[TD];

  const int b    = blockIdx.x;
  const int tid  = threadIdx.x;
  const int lane = tid & 31;
  const int wave = tid >> 5;
  const int lm   = lane & 15;           // m (A) / n (B,C,D)
  const int lk   = (lane >> 4) << 1;    // k-pair base: 0 or 2
  const int hi   = (lane >> 4) << 3;    // C/D row offset: 0 or 8

  float* tok = tok_ws + (size_t)b * TT * TD;

  // ---- build token sequence: [xs|ys] rows + [query_x|0] row ----
  for (int idx = tid; idx < TT * TD; idx += 256) {
    int t = idx >> 7, c = idx & 127;
    float v;
    if (t < NTRAIN) v = (c < DD) ? xs[((size_t)b * NTRAIN + t) * DD + c]
                                 : ys[(size_t)b * NTRAIN + t];
    else            v = (c < DD) ? qx[(size_t)b * DD + c] : 0.0f;
    tok[idx] = v;
  }
  __threadfence();
  __syncthreads();

  for (int layer = 0; layer < NLAYER; ++layer) {
    const float* wq = Wq  + (size_t)layer * DD * TD;   // 127 x 128 (row-major)
    const float* wk = Wk  + (size_t)layer * DD * TD;
    const float* wv = Wv  + (size_t)layer * TD * TD;
    const float* wg = Wgw + (size_t)layer * TD * TD;
    const float* wo = Wo  + (size_t)layer * TD * TD;

    for (int i = tid; i < TD; i += 256) {
      sL1w[i] = Ln1w[layer * TD + i]; sL1b[i] = Ln1b[layer * TD + i];
      sL2w[i] = Ln2w[layer * TD + i]; sL2b[i] = Ln2b[layer * TD + i];
      sGb[i]  = Wgb [layer * TD + i];
      sSk[i]  = 0.0f;
    }
    for (int i = tid; i < TD * TD; i += 256) sS[i] = 0.0f;
    __syncthreads();

    // prime the pipeline: stage chunk 0 into buffer 0
    stage_chunk(tok, sXc0, tid);

    for (int ch = 0; ch < NCHUNK; ++ch) {
      float* xc  = (ch & 1) ? sXc1 : sXc0;          // this chunk's tokens
      float* xcn = (ch & 1) ? sXc0 : sXc1;          // next chunk's buffer
#if USE_ASYNC_LDS
      __builtin_amdgcn_s_wait_asynccnt(0);          // chunk ch DMA complete
#endif
      __syncthreads();

      // ---- LN1: 2 rows per wave, lane-strided + shuffle reduce ----
      for (int rr = 0; rr < 2; ++rr) {
        int r = wave * 2 + rr;
        float s1 = 0.f, s2 = 0.f;
        for (int j = lane; j < TD; j += 32) {
          float x = xc[r * TD + j]; s1 += x; s2 += x * x;
        }
        for (int m = 16; m; m >>= 1) {
          s1 += __shfl_xor(s1, m, 32); s2 += __shfl_xor(s2, m, 32);
        }
        float mean = s1 * (1.0f / TD);
        float var  = s2 * (1.0f / TD) - mean * mean;
        float rstd = rsqrtf(var + LN_EPS);
        for (int j = lane; j < TD; j += 32) {
          float x = xc[r * TD + j];
          sXn[r * TD + j] = (x - mean) * rstd * sL1w[j] + sL1b[j];
        }
      }
      __syncthreads();

      // ---- stage chunk ch+1 while this chunk computes (double buffer) ----
      if (ch + 1 < NCHUNK) {
        const float* tnext = tok + (size_t)(ch + 1) * CH * TD;
        stage_chunk(tnext, xcn, tid);
#if !USE_ASYNC_LDS
        if (ch + 2 < NCHUNK)
          __builtin_prefetch(tnext + CH * TD + tid * 8, 0, 0);
#endif
      }

      // ---- projections: wave w owns output columns [16w,16w+16) ----
      {
        const int col = wave * 16 + lm;          // B-operand row = out feature
        const int rqk = (col < DD) ? col : (DD - 1); // clamp pad row in-bounds
        v8f cq = {}, ck2 = {}, cv = {}, cg = {};
#pragma unroll 4
        for (int kk = 0; kk < 32; ++kk) {
          int k0 = kk * 4 + lk;
          v2f a;  a.x  = sXn[lm * TD + k0];  a.y  = sXn[lm * TD + k0 + 1];
          v2f bq; bq.x = wq[rqk * TD + k0];  bq.y = wq[rqk * TD + k0 + 1];
          v2f bk; bk.x = wk[rqk * TD + k0];  bk.y = wk[rqk * TD + k0 + 1];
          if (col >= DD) { bq.x = 0.f; bq.y = 0.f; bk.x = 0.f; bk.y = 0.f; }
          v2f bv; bv.x = wv[col * TD + k0];  bv.y = wv[col * TD + k0 + 1];
          v2f bg; bg.x = wg[col * TD + k0];  bg.y = wg[col * TD + k0 + 1];
          cq  = wmma_f32(a, bq, cq);
          ck2 = wmma_f32(a, bk, ck2);
          cv  = wmma_f32(a, bv, cv);
          cg  = wmma_f32(a, bg, cg);
        }
        const float bias = sGb[col];
#pragma unroll
        for (int r = 0; r < 8; ++r) {
          int m = r + hi;
          float q = cq[r], k = ck2[r], v = cv[r], g = cg[r];
          // phi(x)=elu(x)+1 = x>0 ? x+1 : exp(x); padded feature -> 0
          float qp = (col < DD) ? (q > 0.f ? q + 1.f : __expf(q)) : 0.f;
          float kp = (col < DD) ? (k > 0.f ? k + 1.f : __expf(k)) : 0.f;
          float gg = 1.0f / (1.0f + __expf(-(g + bias)));
          sQp[m * TD + col] = qp;
          sKp[m * TD + col] = kp;
          sVg[m * TD + col] = v * gg;
        }
      }
      __syncthreads();

      // ---- inter-chunk numerator: Y1 = Qp @ S (per-wave 16x16 tile) ----
      v8f acc = {};
#pragma unroll 4
      for (int kk = 0; kk < 32; ++kk) {
        int k0 = kk * 4 + lk;
        v2f a;  a.x = sQp[lm * TD + k0];         a.y = sQp[lm * TD + k0 + 1];
        v2f bb; bb.x = sS[k0 * TD + wave * 16 + lm];
                bb.y = sS[(k0 + 1) * TD + wave * 16 + lm];
        acc = wmma_f32(a, bb, acc);
      }
      // ---- wave 0: A = Qp @ Kp^T, causal mask ----
      if (wave == 0) {
        v8f ca = {};
#pragma unroll 4
        for (int kk = 0; kk < 32; ++kk) {
          int k0 = kk * 4 + lk;
          v2f a;  a.x  = sQp[lm * TD + k0]; a.y  = sQp[lm * TD + k0 + 1];
          v2f bb; bb.x = sKp[lm * TD + k0]; bb.y = sKp[lm * TD + k0 + 1];
          ca = wmma_f32(a, bb, ca);
        }
#pragma unroll
        for (int r = 0; r < 8; ++r) {
          int m = r + hi;
          sA[m * CH + lm] = (lm <= m) ? ca[r] : 0.0f;
        }
      }
      __syncthreads();

      // ---- denominator: Qp_t . S_k (parallel partials) + rowsum(masked A) --
      {
        int t = tid >> 4, g = tid & 15;
        float p = 0.f;
#pragma unroll
        for (int j = 0; j < 8; ++j) p += sQp[t * TD + g * 8 + j] * sSk[g * 8 + j];
        sDp[t * 16 + g] = p;
      }
      __syncthreads();
      if (tid < CH) {
        float dsum = 0.f;
        for (int g = 0; g < 16; ++g) dsum += sDp[tid * 16 + g];
        for (int s = 0; s <= tid; ++s) dsum += sA[tid * CH + s];
        sDen[tid] = fmaxf(dsum, EPSV);
      }
      __syncthreads();

      // ---- intra-chunk numerator: acc += maskedA @ Vg tile ----
#pragma unroll
      for (int kk = 0; kk < 4; ++kk) {
        int k0 = kk * 4 + lk;
        v2f a;  a.x = sA[lm * CH + k0];          a.y = sA[lm * CH + k0 + 1];
        v2f bb; bb.x = sVg[k0 * TD + wave * 16 + lm];
                bb.y = sVg[(k0 + 1) * TD + wave * 16 + lm];
        acc = wmma_f32(a, bb, acc);
      }
#pragma unroll
      for (int r = 0; r < 8; ++r) {
        int m = r + hi;
        sAt[m * TD + wave * 16 + lm] = acc[r] / sDen[m];
      }
      __syncthreads();

      // ---- state update: S += Kp^T @ Vg (wave owns H-rows [16w,16w+16)) ----
      for (int nt = 0; nt < 8; ++nt) {
        v8f cs;
#pragma unroll
        for (int r = 0; r < 8; ++r)
          cs[r] = sS[(wave * 16 + r + hi) * TD + nt * 16 + lm];
#pragma unroll
        for (int kk = 0; kk < 4; ++kk) {
          int k0 = kk * 4 + lk;
          v2f a;  a.x = sKp[k0 * TD + wave * 16 + lm];
                  a.y = sKp[(k0 + 1) * TD + wave * 16 + lm];
          v2f bb; bb.x = sVg[k0 * TD + nt * 16 + lm];
                  bb.y = sVg[(k0 + 1) * TD + nt * 16 + lm];
          cs = wmma_f32(a, bb, cs);
        }
#pragma unroll
        for (int r = 0; r < 8; ++r)
          sS[(wave * 16 + r + hi) * TD + nt * 16 + lm] = cs[r];
      }
      if (tid < TD) {
        float s = sSk[tid];
        for (int k = 0; k < CH; ++k) s += sKp[k * TD + tid];
        sSk[tid] = s;
      }
      __syncthreads();

      // ---- output projection + residual: out = Xc + 0.1 * attn @ Wo^T ----
      {
        v8f co = {};
        const int row = wave * 16 + lm;
#pragma unroll 4
        for (int kk = 0; kk < 32; ++kk) {
          int k0 = kk * 4 + lk;
          v2f a;  a.x  = sAt[lm * TD + k0]; a.y  = sAt[lm * TD + k0 + 1];
          v2f bb; bb.x = wo[row * TD + k0]; bb.y = wo[row * TD + k0 + 1];
          co = wmma_f32(a, bb, co);
        }
#pragma unroll
        for (int r = 0; r < 8; ++r) {
          int m = r + hi;
          sOu[m * TD + row] = xc[m * TD + row] + 0.1f * co[r];
        }
      }
      __syncthreads();

      // ---- LN2 -> write chunk back to token buffer (in-place) ----
      for (int rr = 0; rr < 2; ++rr) {
        int r = wave * 2 + rr;
        float s1 = 0.f, s2 = 0.f;
        for (int j = lane; j < TD; j += 32) {
          float x = sOu[r * TD + j]; s1 += x; s2 += x * x;
        }
        for (int m = 16; m; m >>= 1) {
          s1 += __shfl_xor(s1, m, 32); s2 += __shfl_xor(s2, m, 32);
        }
        float mean = s1 * (1.0f / TD);
        float var  = s2 * (1.0f / TD) - mean * mean;
        float rstd = rsqrtf(var + LN_EPS);
        for (int j = lane; j < TD; j += 32) {
          float y = (sOu[r * TD + j] - mean) * rstd * sL2w[j] + sL2b[j];
          tok[((size_t)ch * CH + r) * TD + j] = y;
          if (layer == NLAYER - 1 && ch == NCHUNK - 1 && r == CH - 1)
            sFin[j] = y;
        }
      }
      __threadfence();
      __syncthreads();
    } // chunk
    __syncthreads();
  } // layer

  if (tid == 0) {
    float y = 0.f;
    for (int j = 0; j < TD; ++j) y += sFin[j] * predw[j];
    out[b] = y;
  }
}

extern "C" void kernel_launch(void* const* d_in, const int* in_sizes, int n_in,
                              void* d_out, int out_size, void* d_ws, size_t ws_size,
                              hipStream_t stream) {
  const float* xs    = (const float*)d_in[0];
  const float* ys    = (const float*)d_in[1];
  const float* qx    = (const float*)d_in[2];
  const float* Wq    = (const float*)d_in[3];
  const float* Wk    = (const float*)d_in[4];
  const float* Wv    = (const float*)d_in[5];
  const float* Wgw   = (const float*)d_in[6];
  const float* Wgb   = (const float*)d_in[7];
  const float* Wo    = (const float*)d_in[8];
  const float* ln1w  = (const float*)d_in[9];
  const float* ln1b  = (const float*)d_in[10];
  const float* ln2w  = (const float*)d_in[11];
  const float* ln2b  = (const float*)d_in[12];
  const float* predw = (const float*)d_in[13];
  float* out    = (float*)d_out;
  float* tok_ws = (float*)d_ws;   // needs 2*4096*128*4 = 4 MB

  gla_fused_kernel<<<dim3(2), dim3(256), 0, stream>>>(
      xs, ys, qx, Wq, Wk, Wv, Wgw, Wgb, Wo,
      ln1w, ln1b, ln2w, ln2b, predw, out, tok_ws);
}